// BERTCRFForTokenClassification_74328704024639
// MI455X (gfx1250) — compile-verified
//
#include <hip/hip_runtime.h>

typedef __attribute__((ext_vector_type(2))) float v2f;
typedef __attribute__((ext_vector_type(8))) float v8f;

#define BT 32768   // B*T rows of the GEMM
#define HD 768     // K
#define NP (HD/2)  // K-pairs
#define LL 21      // labels
#define TT 512
#define BB 64
#define NEG_INF (-1e30f)

// ---------------------------------------------------------------------------
// Kernel 1: emissions = hidden @ W + b  via V_WMMA_F32_16X16X4_F32
// One wave per 16-row M tile; two v8f accumulators cover N=0..31.
// A-frag (ISA 7.12.2, 32-bit A 16x4): lane<16 -> K={0,1}, lane>=16 -> K={2,3},
// i.e. a float2 load at hidden[row*768 + k0 + 2*(lane>>4)].
// B-frags come from W restaged in LDS as K-pair-interleaved float2
// (Wp[k/2][n] = {W[k][n], W[k+1][n]}), so each operand is one aligned
// ds_load_b64 straight into an even VGPR pair -- no re-pairing moves.
// Columns 21..31 of acc1 read the next K-pair's Wp entries (the zero pad on the last) and are never stored.
// ---------------------------------------------------------------------------
__global__ __launch_bounds__(256) void emis_gemm(const float* __restrict__ hidden,
                                                 const float* __restrict__ W,
                                                 const float* __restrict__ bias,
                                                 float* __restrict__ emis) {
  __shared__ v2f Wp[NP * LL + 16];   // 64.6 KB; pad zero-filled
  const int tid = threadIdx.x;
  for (int idx = tid; idx < NP * LL + 16; idx += 256) {
    v2f w = {0.0f, 0.0f};
    if (idx < NP * LL) {
      const int p = idx / LL;
      const int n = idx % LL;
      w.x = W[(2 * p) * LL + n];
      w.y = W[(2 * p + 1) * LL + n];
    }
    Wp[idx] = w;
  }
  __syncthreads();

  const int lane  = tid & 31;
  const int wave  = tid >> 5;
  const int mtile = blockIdx.x * 8 + wave;
  const int mrow  = (mtile << 4) + (lane & 15);
  const int koff  = (lane >> 4) << 1;       // 0 or 2
  const int n0    = lane & 15;              // cols 0..15
  const int n1    = n0 + 16;                // cols 16..31 (stored only if < 21)
  const bool n1ok = (n1 < LL);
  const float* hrow = hidden + (size_t)mrow * HD + koff;

  v8f acc0 = {};
  v8f acc1 = {};
  #pragma unroll 8
  for (int k0 = 0; k0 < HD; k0 += 4) {
    v2f a = *(const v2f*)(hrow + k0);                 // global_load_b64
    const int pb = ((k0 + koff) >> 1) * LL + n0;      // K-pair row
    v2f b0 = Wp[pb];                                  // ds_load_b64
    v2f b1 = Wp[pb + 16];                             // ds_load_b64 (+128B)
    acc0 = __builtin_amdgcn_wmma_f32_16x16x4_f32(false, a, false, b0, (short)0, acc0, false, false);
    acc1 = __builtin_amdgcn_wmma_f32_16x16x4_f32(false, a, false, b1, (short)0, acc1, false, false);
  }

  // C/D layout: VGPR r -> M = r + 8*(lane>>4), N = lane&15 (+16 for acc1)
  const int rbase = (lane >> 4) << 3;
  const float bias0 = bias[n0];
  const float bias1 = n1ok ? bias[n1] : 0.0f;
  #pragma unroll
  for (int r = 0; r < 8; ++r) {
    const int orow = (mtile << 4) + rbase + r;
    emis[(size_t)orow * LL + n0] = acc0[r] + bias0;
    if (n1ok) emis[(size_t)orow * LL + n1] = acc1[r] + bias1;
  }
}

// ---------------------------------------------------------------------------
// Kernel 2: CRF forward (log-partition) + gold-path score, one wave per batch.
// logsumexp_i(a_i + T_ij) = mx + log( sum_i exp(a_i - mx) * exp(T_ij) ),
// exp(T) precomputed into LDS -> inner loop is 21 FMAs per lane per step.
// ---------------------------------------------------------------------------
__global__ __launch_bounds__(32) void crf_nll(const float* __restrict__ emis,
                                              const float* __restrict__ st,
                                              const float* __restrict__ et,
                                              const float* __restrict__ trans,
                                              const int* __restrict__ labels,
                                              float* __restrict__ sb) {
  __shared__ float shE[LL * 22];   // exp(trans[i][j]) at [i*22 + j]
  __shared__ float sh[32];
  const int b = blockIdx.x;
  const int j = threadIdx.x;
  for (int idx = j; idx < LL * LL; idx += 32)
    shE[(idx / LL) * 22 + (idx % LL)] = __expf(trans[idx]);
  __syncthreads();

  const float* em = emis + (size_t)b * TT * LL;
  const int* lab  = labels + (size_t)b * TT;
  const bool act  = (j < LL);

  float alpha = act ? (st[j] + em[j]) : NEG_INF;
  for (int t = 1; t < TT; ++t) {
    float mx = alpha;
    #pragma unroll
    for (int o = 16; o > 0; o >>= 1) mx = fmaxf(mx, __shfl_xor(mx, o, 32));
    const float ea = act ? __expf(alpha - mx) : 0.0f;
    __syncthreads();
    sh[j] = ea;
    __syncthreads();
    if (act && (lab[t] != -100)) {        // mask: keep alpha where m==0
      float s = 0.0f;
      #pragma unroll
      for (int i = 0; i < LL; ++i) s += sh[i] * shE[i * 22 + j];
      alpha = mx + __logf(s) + em[t * LL + j];
    }
  }

  // logZ = logsumexp_j(alpha_j + et_j)
  float v = act ? (alpha + et[j]) : NEG_INF;
  float mx = v;
  #pragma unroll
  for (int o = 16; o > 0; o >>= 1) mx = fmaxf(mx, __shfl_xor(mx, o, 32));
  float s = act ? __expf(v - mx) : 0.0f;
  #pragma unroll
  for (int o = 16; o > 0; o >>= 1) s += __shfl_xor(s, o, 32);
  const float logZ = mx + __logf(s);

  // gold-path numerator, lanes split the T loop
  float part = 0.0f;
  int cnt = 0;
  for (int t = j; t < TT; t += 32) {
    const int l = lab[t];
    if (t == 0) {
      const int tag = (l == -100) ? 0 : l;
      part += st[tag] + em[tag];
      cnt += 1;
    } else if (l != -100) {
      const int lp   = lab[t - 1];
      const int tagp = (lp == -100) ? 0 : lp;
      part += trans[tagp * LL + l] + em[t * LL + l];
      cnt += 1;
    }
  }
  #pragma unroll
  for (int o = 16; o > 0; o >>= 1) {
    part += __shfl_xor(part, o, 32);
    cnt  += __shfl_xor(cnt, o, 32);
  }
  if (j == 0) {
    const int li   = cnt - 1;              // last valid index
    const int lt   = lab[li];
    const int ltag = (lt == -100) ? 0 : lt;
    sb[b] = (part + et[ltag]) - logZ;      // log-likelihood of batch b
  }
}

// ---------------------------------------------------------------------------
// Kernel 3: mean-reduce per-batch log-likelihoods -> nll scalar
// ---------------------------------------------------------------------------
__global__ __launch_bounds__(64) void nll_reduce(const float* __restrict__ sb,
                                                 float* __restrict__ out) {
  __shared__ float sh[64];
  const int t = threadIdx.x;
  sh[t] = sb[t];
  __syncthreads();
  for (int o = 32; o > 0; o >>= 1) {
    if (t < o) sh[t] += sh[t + o];
    __syncthreads();
  }
  if (t == 0) out[0] = -sh[0] / (float)BB;
}

// ---------------------------------------------------------------------------
// Kernel 4: Viterbi, one wave per batch. Back-pointer history kept in LDS
// (511*21 bytes) so the sequential backtrace chases LDS latency, not L2.
// Identity back-pointers at masked steps (matches reference).
// ---------------------------------------------------------------------------
__global__ __launch_bounds__(32) void viterbi(const float* __restrict__ emis,
                                              const float* __restrict__ st,
                                              const float* __restrict__ et,
                                              const float* __restrict__ trans,
                                              const int* __restrict__ attn,
                                              int* __restrict__ path) {
  __shared__ float shT[LL * 22];
  __shared__ float sh[32];
  __shared__ unsigned char hb[(TT - 1) * LL];   // 10731 B
  const int b = blockIdx.x;
  const int j = threadIdx.x;
  for (int idx = j; idx < LL * LL; idx += 32)
    shT[(idx / LL) * 22 + (idx % LL)] = trans[idx];
  __syncthreads();

  const float* em = emis + (size_t)b * TT * LL;
  const int* am   = attn + (size_t)b * TT;
  const bool act  = (j < LL);

  float score = act ? (st[j] + em[j]) : NEG_INF;
  for (int t = 1; t < TT; ++t) {
    __syncthreads();
    sh[j] = score;
    __syncthreads();
    const bool m = (am[t] != 0);
    if (act) {
      int idx = j;                           // identity when masked
      if (m) {
        float best = sh[0] + shT[j];
        int bi = 0;
        #pragma unroll
        for (int i = 1; i < LL; ++i) {
          const float v = sh[i] + shT[i * 22 + j];
          if (v > best) { best = v; bi = i; }   // strict > keeps first max
        }
        score = best + em[t * LL + j];
        idx = bi;
      }
      hb[(t - 1) * LL + j] = (unsigned char)idx;
    }
  }

  __syncthreads();
  sh[j] = act ? (score + et[j]) : NEG_INF;
  __syncthreads();
  if (j == 0) {
    float best = sh[0];
    int bi = 0;
    for (int i = 1; i < LL; ++i)
      if (sh[i] > best) { best = sh[i]; bi = i; }
    int tag = bi;
    int* pb = path + (size_t)b * TT;
    pb[TT - 1] = tag;
    for (int t = TT - 2; t >= 0; --t) {
      tag = hb[t * LL + tag];
      pb[t] = tag;
    }
  }
}

// ---------------------------------------------------------------------------
// Kernel 5: logits = one_hot(path) * attention_mask
// ---------------------------------------------------------------------------
__global__ __launch_bounds__(256) void fill_logits(const int* __restrict__ path,
                                                   const int* __restrict__ attn,
                                                   float* __restrict__ logits) {
  const int gid = blockIdx.x * blockDim.x + threadIdx.x;   // 0 .. B*T-1
  if (gid >= BB * TT) return;
  const int m = attn[gid];
  const int p = path[gid];
  float* o = logits + (size_t)gid * LL;
  #pragma unroll
  for (int l = 0; l < LL; ++l) o[l] = (m != 0 && l == p) ? 1.0f : 0.0f;
}

// ---------------------------------------------------------------------------
extern "C" void kernel_launch(void* const* d_in, const int* in_sizes, int n_in,
                              void* d_out, int out_size, void* d_ws, size_t ws_size,
                              hipStream_t stream) {
  const float* hidden = (const float*)d_in[0];   // (64,512,768)
  const float* W      = (const float*)d_in[1];   // (768,21)
  const float* bias   = (const float*)d_in[2];   // (21)
  const float* st     = (const float*)d_in[3];   // (21)
  const float* et     = (const float*)d_in[4];   // (21)
  const float* trans  = (const float*)d_in[5];   // (21,21)
  const int*   labels = (const int*)d_in[6];     // (64,512) int32
  const int*   attn   = (const int*)d_in[7];     // (64,512) int32

  float* out    = (float*)d_out;
  float* nll    = out;                     // [0]
  float* logits = out + 1;                 // 64*512*21
  float* emis   = out + 1 + BB * TT * LL;  // 64*512*21

  // workspace: path (B*T ints = 128 KB) + per-batch scores (64 floats)
  int*   path = (int*)d_ws;
  float* sb   = (float*)((char*)d_ws + (size_t)BB * TT * sizeof(int));

  emis_gemm<<<BT / (16 * 8), 256, 0, stream>>>(hidden, W, bias, emis);
  crf_nll<<<BB, 32, 0, stream>>>(emis, st, et, trans, labels, sb);
  nll_reduce<<<1, 64, 0, stream>>>(sb, nll);
  viterbi<<<BB, 32, 0, stream>>>(emis, st, et, trans, attn, path);
  fill_logits<<<(BB * TT + 255) / 256, 256, 0, stream>>>(path, attn, logits);
}